// WebSwarmLDA_26542897889838
// MI455X (gfx1250) — compile-verified
//
#include <hip/hip_runtime.h>
#include <hip/hip_bf16.h>

// ---------------------------------------------------------------------------
// Problem constants (from the reference)
// ---------------------------------------------------------------------------
#define B_ROWS   2048
#define D_COLS   20000
#define NEXP     1000
#define G_GENES  128
#define NC       10     // C
#define NH1      10
#define NH2      20
#define NH3      10
#define NWAVES   16     // 512 threads / wave32
#define BN_EPS   1e-5f

typedef __attribute__((ext_vector_type(2))) float        v2f;
typedef __attribute__((ext_vector_type(8))) float        v8f;
typedef __attribute__((ext_vector_type(4))) unsigned int v4u;
typedef __attribute__((ext_vector_type(4))) int          v4i;
typedef __attribute__((ext_vector_type(8))) int          v8i;

// fp32 WMMA: D(16x16) = A(16x4) * B(4x16) + C   (native fp32 matrix op on CDNA5)
__device__ __forceinline__ v8f wmma4(v2f a, v2f b, v8f c) {
  return __builtin_amdgcn_wmma_f32_16x16x4_f32(
      /*neg_a=*/false, a, /*neg_b=*/false, b,
      /*c_mod=*/(short)0, c, /*reuse_a=*/false, /*reuse_b=*/false);
}

// ---------------------------------------------------------------------------
// Tensor Data Mover: 1-D DMA of `n4` 4-byte elements global -> LDS.
// D# built per CDNA5 ISA (group0 + group1; groups 2/3 zero,
// 2-D max). LDS offset = low 32 bits of the generic pointer (aperture rule).
// Issue from ONE wave only (TDM ignores EXEC; per-wave issue).
// ---------------------------------------------------------------------------
__device__ __forceinline__ void tdm_load_1d(void* lds_dst, const void* gsrc,
                                            unsigned n4) {
  unsigned lds_off = (unsigned)(unsigned long long)(uintptr_t)lds_dst;
  unsigned long long ga = (unsigned long long)(uintptr_t)gsrc;
  v4u g0;
  g0.x = 1u;                                             // count=1 (valid D#)
  g0.y = lds_off;                                        // lds_addr
  g0.z = (unsigned)(ga & 0xFFFFFFFFu);                   // global_addr[31:0]
  g0.w = ((unsigned)(ga >> 32) & 0x01FFFFFFu)            // global_addr[56:32]
         | 0x80000000u;                                  // type=2 ("image")
  v8i g1;
  g1[0] = (int)(2u << 16);                               // data_size=4B, mask=0
  g1[1] = (int)((n4 & 0xFFFFu) << 16);                   // tensor_dim0[15:0]
  g1[2] = (int)(((n4 >> 16) & 0xFFFFu) | (1u << 16));    // dim0 hi | tensor_dim1=1
  g1[3] = (int)((n4 & 0xFFFFu) << 16);                   // tile_dim0 = n4
  g1[4] = 1;                                             // tile_dim1=1, tile_dim2=0
  g1[5] = (int)n4;                                       // tensor_dim0_stride lo32
  g1[6] = 0;                                             // stride hi | dim1_stride lo
  g1[7] = 0;
  v4i z4 = {0, 0, 0, 0};
#if __clang_major__ >= 23
  v8i z8 = {0, 0, 0, 0, 0, 0, 0, 0};
  __builtin_amdgcn_tensor_load_to_lds(g0, g1, z4, z4, z8, 0);
#else
  __builtin_amdgcn_tensor_load_to_lds(g0, g1, z4, z4, 0);
#endif
}

// A-fragment element for MLP layers: BN scale/shift (+ optional ReLU) folded in.
template <int CIN, bool RELU>
__device__ __forceinline__ float load_in(const float* __restrict__ aRow, int k,
                                         const float* __restrict__ scl,
                                         const float* __restrict__ shf) {
  if (k >= CIN) return 0.0f;                 // zero K-padding
  float x = aRow[k];
  float y = fmaf(x, scl[k], shf[k]);
  return RELU ? fmaxf(y, 0.0f) : y;
}

// Stage transposed, zero-padded weights into LDS: dst[k*npad+n] = w[n*cin+k]
__device__ __forceinline__ void fill_w(float* __restrict__ dst,
                                       const float* __restrict__ w,
                                       int cout, int cin, int kpad, int npad,
                                       int tid, int bd) {
  for (int i = tid; i < kpad * npad; i += bd) {
    int k = i / npad, n = i % npad;
    dst[i] = (k < cin && n < cout) ? w[n * cin + k] : 0.0f;
  }
}

__device__ __forceinline__ void fill_bias(float* __restrict__ dst,
                                          const float* __restrict__ b,
                                          int cout, int npad, int tid, int bd) {
  for (int i = tid; i < npad; i += bd) dst[i] = (i < cout) ? b[i] : 0.0f;
}

// Reduce LDS sums -> BN scale/shift, and re-zero sums for the next layer.
__device__ __forceinline__ void bn_update(int tid, int e, int cout,
                                          const float* __restrict__ bng,
                                          const float* __restrict__ bnb,
                                          float* __restrict__ sums,
                                          float* __restrict__ scl,
                                          float* __restrict__ shf) {
  __syncthreads();
  if (tid < 32) {
    float s1 = sums[tid], s2 = sums[32 + tid];
    sums[tid] = 0.0f;
    sums[32 + tid] = 0.0f;
    if (tid < cout) {
      const float inv = 1.0f / (float)B_ROWS;
      float m   = s1 * inv;
      float var = fmaf(-m, m, s2 * inv);
      float rs  = rsqrtf(var + BN_EPS);
      float g   = bng[(size_t)e * cout + tid];
      float bb  = bnb[(size_t)e * cout + tid];
      scl[tid] = rs * g;
      shf[tid] = fmaf(-m, rs * g, bb);   // b - m*rs*g
    }
  }
  __syncthreads();
}

// One MLP layer as a WMMA GEMM over the whole batch.
// actOut gets raw (linear + bias); BN/ReLU are applied lazily on the next read.
template <int CIN, int KPAD, int COUT, int NPAD, bool RELU_IN, bool FINAL>
__device__ __forceinline__ void mlp_layer(
    const float* __restrict__ actIn, int strideIn,
    float* __restrict__ actOut, int strideOut,
    const float* __restrict__ Bw, const float* __restrict__ bias,
    const float* __restrict__ scl, const float* __restrict__ shf,
    float* __restrict__ sums, float* __restrict__ gout, float aggw,
    int wave, int laneN, int laneHi) {
  for (int tile = wave; tile < B_ROWS / 16; tile += NWAVES) {
    const int rowbase = tile * 16;
    const float* aRow = actIn + (size_t)(rowbase + laneN) * strideIn;
#pragma unroll
    for (int nt = 0; nt < NPAD / 16; ++nt) {
      v8f acc = {};
#pragma unroll
      for (int kk = 0; kk < KPAD; kk += 4) {
        const int k0 = kk + 2 * laneHi;
        v2f a, b;
        a.x = load_in<CIN, RELU_IN>(aRow, k0, scl, shf);
        a.y = load_in<CIN, RELU_IN>(aRow, k0 + 1, scl, shf);
        b.x = Bw[k0 * NPAD + nt * 16 + laneN];
        b.y = Bw[(k0 + 1) * NPAD + nt * 16 + laneN];
        acc = wmma4(a, b, acc);
      }
      const int col = nt * 16 + laneN;
      const bool valid = col < COUT;
      const float bi = valid ? bias[col] : 0.0f;
      float s1 = 0.0f, s2 = 0.0f;
#pragma unroll
      for (int r = 0; r < 8; ++r) {
        const float v = acc[r] + bi;
        const int row = rowbase + r + 8 * laneHi;
        if (valid) {
          if constexpr (FINAL) {
            atomicAdd(&gout[row * NC + col], aggw * v);   // global_atomic_add_f32
          } else {
            actOut[(size_t)row * strideOut + col] = v;
            s1 += v;
            s2 += v * v;
          }
        }
      }
      if constexpr (!FINAL) {
        if (valid) {
          atomicAdd(&sums[col], s1);        // ds_add_f32
          atomicAdd(&sums[32 + col], s2);
        }
      }
    }
  }
}

// ---------------------------------------------------------------------------
// Kernels
// ---------------------------------------------------------------------------
__global__ void WebSwarmLDA_init_out(float* __restrict__ out,
                                     const float* __restrict__ aggb, int n) {
  int i = blockIdx.x * blockDim.x + threadIdx.x;
  if (i < n) out[i] = aggb[0];
}

// LDS layout (floats), total 66208 floats = 264832 bytes (< 320KB WGP budget)
#define SMEM_FLOATS 66208
#define SMEM_BYTES  (SMEM_FLOATS * 4)

__global__ void __launch_bounds__(512, 1) WebSwarmLDA_expert_kernel(
    const float* __restrict__ X, const int* __restrict__ genes,
    const float* __restrict__ lda_w, const float* __restrict__ lda_b,
    const float* __restrict__ w1, const float* __restrict__ b1,
    const float* __restrict__ w2, const float* __restrict__ b2,
    const float* __restrict__ w3, const float* __restrict__ b3,
    const float* __restrict__ w4, const float* __restrict__ b4,
    const float* __restrict__ bn0g, const float* __restrict__ bn0b,
    const float* __restrict__ bn1g, const float* __restrict__ bn1b,
    const float* __restrict__ bn2g, const float* __restrict__ bn2b,
    const float* __restrict__ bn3g, const float* __restrict__ bn3b,
    const float* __restrict__ aggW, float* __restrict__ out) {
  extern __shared__ float smemf[];

  float* Blda    = smemf;            // 128*16  = 2048 (transposed, padded)
  float* Bw1p    = Blda + 2048;      // 12*16   = 192
  float* Bw2p    = Bw1p + 192;       // 12*32   = 384
  float* Bw3p    = Bw2p + 384;       // 20*16   = 320
  float* Bw4p    = Bw3p + 320;       // 12*16   = 192
  float* bias0   = Bw4p + 192;       // 16
  float* bias1   = bias0 + 16;       // 16
  float* bias2   = bias1 + 16;       // 32
  float* bias3   = bias2 + 32;       // 16
  float* bias4   = bias3 + 16;       // 16
  float* scl     = bias4 + 16;       // 32
  float* shf     = scl + 32;         // 32
  float* sums    = shf + 32;         // 64 (sum | sumsq)
  int*   gidx    = (int*)(sums + 64);      // 128
  float* actA    = (float*)(gidx + 128);   // 2048*20 = 40960 (h0 s10, then h2 s20)
  float* actB    = actA + 40960;           // 2048*10 = 20480 (h1, then h3)
  float* BldaRaw = actB + 20480;           // 10*128  = 1280 (TDM staging)

  const int tid = threadIdx.x;
  const int e   = blockIdx.x;
  const int bd  = 512;

  // ---- phase 1: TDM the LDA weight block; stage everything else ----------
  if (tid < 32) {   // one wave issues the DMA (TDM ignores EXEC; per-wave op)
    tdm_load_1d(BldaRaw, lda_w + (size_t)e * NC * G_GENES, NC * G_GENES);
  }
  for (int i = tid; i < G_GENES; i += bd)
    gidx[i] = genes[(size_t)e * G_GENES + i];
  fill_w(Bw1p, w1 + (size_t)e * NH1 * NC,  NH1, NC,  12, 16, tid, bd);
  fill_w(Bw2p, w2 + (size_t)e * NH2 * NH1, NH2, NH1, 12, 32, tid, bd);
  fill_w(Bw3p, w3 + (size_t)e * NH3 * NH2, NH3, NH2, 20, 16, tid, bd);
  fill_w(Bw4p, w4 + (size_t)e * NC * NH3,  NC,  NH3, 12, 16, tid, bd);
  fill_bias(bias0, lda_b + (size_t)e * NC,  NC,  16, tid, bd);
  fill_bias(bias1, b1 + (size_t)e * NH1,    NH1, 16, tid, bd);
  fill_bias(bias2, b2 + (size_t)e * NH2,    NH2, 32, tid, bd);
  fill_bias(bias3, b3 + (size_t)e * NH3,    NH3, 16, tid, bd);
  fill_bias(bias4, b4 + (size_t)e * NC,     NC,  16, tid, bd);
  if (tid < 64) sums[tid] = 0.0f;
  if (tid < 32) {
    __builtin_amdgcn_s_wait_tensorcnt(0);   // TDM data visible in LDS
  }
  __syncthreads();

  // ---- phase 2: LDS->LDS transpose into padded B layout (bank-friendly) --
  for (int i = tid; i < G_GENES * 16; i += bd) {
    int k = i >> 4, n = i & 15;
    Blda[i] = (n < NC) ? BldaRaw[n * G_GENES + k] : 0.0f;   // B = lda_w^T
  }
  __syncthreads();

  const int wave   = tid >> 5;
  const int lane   = tid & 31;
  const int laneN  = lane & 15;   // M (for A) / N (for B,D)
  const int laneHi = lane >> 4;   // selects K {0,1} vs {2,3}; D rows +8
  const float aggw = aggW[e];

  // ---- LDA layer: h0 = gather(X) @ lda_w^T + lda_b -----------------------
  for (int tile = wave; tile < B_ROWS / 16; tile += NWAVES) {
    const int rowbase = tile * 16;
    const float* Xr = X + (size_t)(rowbase + laneN) * D_COLS;

    // prefetch a slice of next tile's gather addresses (global_prefetch_b8)
    if (tile + NWAVES < B_ROWS / 16) {
      const float* Xp = X + (size_t)((tile + NWAVES) * 16 + laneN) * D_COLS;
#pragma unroll
      for (int k = 0; k < G_GENES; k += 16)
        __builtin_prefetch(&Xp[gidx[k + 8 * laneHi]], 0, 0);
    }

    v8f acc = {};
#pragma unroll 8
    for (int kk = 0; kk < G_GENES; kk += 4) {
      const int k0 = kk + 2 * laneHi;
      v2f a, b;
      a.x = Xr[gidx[k0]];           // random-column gather (L2 resident)
      a.y = Xr[gidx[k0 + 1]];
      b.x = Blda[k0 * 16 + laneN];
      b.y = Blda[(k0 + 1) * 16 + laneN];
      acc = wmma4(a, b, acc);
    }

    const bool valid = laneN < NC;
    const float bi = valid ? bias0[laneN] : 0.0f;
    float s1 = 0.0f, s2 = 0.0f;
#pragma unroll
    for (int r = 0; r < 8; ++r) {
      float v = acc[r] + bi;
      int row = rowbase + r + 8 * laneHi;
      if (valid) {
        actA[row * NC + laneN] = v;
        s1 += v;
        s2 += v * v;
      }
    }
    if (valid) {
      atomicAdd(&sums[laneN], s1);
      atomicAdd(&sums[32 + laneN], s2);
    }
  }

  // ---- BN0 + MLP chain (BN/ReLU folded into the next layer's A-reads) ----
  bn_update(tid, e, NC, bn0g, bn0b, sums, scl, shf);
  mlp_layer<NC, 12, NH1, 16, /*relu_in=*/false, /*final=*/false>(
      actA, NC, actB, NH1, Bw1p, bias1, scl, shf, sums, nullptr, 0.0f,
      wave, laneN, laneHi);

  bn_update(tid, e, NH1, bn1g, bn1b, sums, scl, shf);
  mlp_layer<NH1, 12, NH2, 32, true, false>(
      actB, NH1, actA, NH2, Bw2p, bias2, scl, shf, sums, nullptr, 0.0f,
      wave, laneN, laneHi);

  bn_update(tid, e, NH2, bn2g, bn2b, sums, scl, shf);
  mlp_layer<NH2, 20, NH3, 16, true, false>(
      actA, NH2, actB, NH3, Bw3p, bias3, scl, shf, sums, nullptr, 0.0f,
      wave, laneN, laneHi);

  bn_update(tid, e, NH3, bn3g, bn3b, sums, scl, shf);
  mlp_layer<NH3, 12, NC, 16, true, /*final=*/true>(
      actB, NH3, nullptr, 0, Bw4p, bias4, scl, shf, sums, out, aggw,
      wave, laneN, laneHi);
}

// ---------------------------------------------------------------------------
// Launch
// ---------------------------------------------------------------------------
extern "C" void kernel_launch(void* const* d_in, const int* in_sizes, int n_in,
                              void* d_out, int out_size, void* d_ws, size_t ws_size,
                              hipStream_t stream) {
  const float* X     = (const float*)d_in[0];
  const int*   genes = (const int*)d_in[1];
  const float* ldaw  = (const float*)d_in[2];
  const float* ldab  = (const float*)d_in[3];
  const float* w1    = (const float*)d_in[4];
  const float* b1    = (const float*)d_in[5];
  const float* w2    = (const float*)d_in[6];
  const float* b2    = (const float*)d_in[7];
  const float* w3    = (const float*)d_in[8];
  const float* b3    = (const float*)d_in[9];
  const float* w4    = (const float*)d_in[10];
  const float* b4    = (const float*)d_in[11];
  const float* bn0g  = (const float*)d_in[12];
  const float* bn0b  = (const float*)d_in[13];
  const float* bn1g  = (const float*)d_in[14];
  const float* bn1b  = (const float*)d_in[15];
  const float* bn2g  = (const float*)d_in[16];
  const float* bn2b  = (const float*)d_in[17];
  const float* bn3g  = (const float*)d_in[18];
  const float* bn3b  = (const float*)d_in[19];
  const float* aggW  = (const float*)d_in[20];
  const float* aggB  = (const float*)d_in[21];
  float* out = (float*)d_out;

  // Allow > 64KB dynamic LDS for the expert kernel (WGP has 320KB).
  static bool attr_set = [] {
    hipFuncSetAttribute((const void*)WebSwarmLDA_expert_kernel,
                        hipFuncAttributeMaxDynamicSharedMemorySize, SMEM_BYTES);
    return true;
  }();
  (void)attr_set;

  const int n_out = B_ROWS * NC;  // 20480
  WebSwarmLDA_init_out<<<(n_out + 255) / 256, 256, 0, stream>>>(out, aggB, n_out);
  WebSwarmLDA_expert_kernel<<<NEXP, 512, SMEM_BYTES, stream>>>(
      X, genes, ldaw, ldab, w1, b1, w2, b2, w3, b3, w4, b4,
      bn0g, bn0b, bn1g, bn1b, bn2g, bn2b, bn3g, bn3b, aggW, out);
}